// CrossframeGlobalAttentionModule_39384850105051
// MI455X (gfx1250) — compile-verified
//
#include <hip/hip_runtime.h>
#include <math.h>

typedef __attribute__((ext_vector_type(2))) float v2f;
typedef __attribute__((ext_vector_type(4))) float f4;
typedef __attribute__((ext_vector_type(8))) float v8f;

#define CCH   64   // channels
#define WAVES 4    // waves per block (128 threads)

__device__ __forceinline__ void lds_fence() {
  // make wave-local LDS store->load visible (CDNA5 split counters)
  asm volatile("s_wait_dscnt 0x0" ::: "memory");
}

// A fragment (16x4 f32): lane l holds A[m=l%16][k = kb + 2*(l/16) + j], j=0,1
__device__ __forceinline__ v2f ldA(const float* t, int lane, int kb) {
  int m = lane & 15;
  int k = kb + ((lane >> 4) << 1);
  return *(const v2f*)(t + m * CCH + k);
}
// B fragment (4x16 f32) for Y = X @ W^T: B[k][n] = W[c0+n][k]
__device__ __forceinline__ v2f ldB(const float* w, int lane, int c0, int kb) {
  int n = lane & 15;
  int k = kb + ((lane >> 4) << 1);
  return *(const v2f*)(w + (c0 + n) * CCH + k);
}

__device__ __forceinline__ void stage_weights(const float* __restrict__ Wh,
                                              const float* __restrict__ Wc,
                                              float* sW1, float* sW2) {
  const f4* g1 = (const f4*)Wh;
  const f4* g2 = (const f4*)Wc;
  f4* s1 = (f4*)sW1;
  f4* s2 = (f4*)sW2;
  for (int i = threadIdx.x; i < (CCH * CCH) / 4; i += blockDim.x) {
    s1[i] = g1[i];
    s2[i] = g2[i];
  }
}

// GEMM1 (X@Wh^T + b) then GEMM2 (relu(T@Wc^T)). Input tile (16x64) in myTile;
// myTile is clobbered (holds T afterwards). Result (relu'd) in y[4] D-fragments.
__device__ __forceinline__ void hidden_forward(float* myTile,
                                               const float* sW1, const float* sW2,
                                               const float* __restrict__ bh,
                                               int lane, v8f (&y)[4]) {
  int n  = lane & 15;
  int mb = (lane >> 4) << 3;

  // ---- GEMM1: T = X @ Wh^T + b_hidden (bias enters via the C operand) ----
  v8f acc[4];
#pragma unroll
  for (int c = 0; c < 4; ++c) {
    float bv = bh[c * 16 + n];
#pragma unroll
    for (int j = 0; j < 8; ++j) acc[c][j] = bv;
  }
#pragma unroll
  for (int kb = 0; kb < CCH; kb += 4) {
    v2f a = ldA(myTile, lane, kb);
#pragma unroll
    for (int c = 0; c < 4; ++c) {
      v2f b = ldB(sW1, lane, c * 16, kb);
      acc[c] = __builtin_amdgcn_wmma_f32_16x16x4_f32(
          false, a, false, b, (short)0, acc[c], false, false);
    }
  }
  // write T back to the wave-private tile (D layout -> row major)
#pragma unroll
  for (int c = 0; c < 4; ++c)
#pragma unroll
    for (int j = 0; j < 8; ++j)
      myTile[(mb + j) * CCH + c * 16 + n] = acc[c][j];
  lds_fence();

  // ---- GEMM2: Y = relu(T @ Wc^T) ----
#pragma unroll
  for (int c = 0; c < 4; ++c)
#pragma unroll
    for (int j = 0; j < 8; ++j) y[c][j] = 0.0f;
#pragma unroll
  for (int kb = 0; kb < CCH; kb += 4) {
    v2f a = ldA(myTile, lane, kb);
#pragma unroll
    for (int c = 0; c < 4; ++c) {
      v2f b = ldB(sW2, lane, c * 16, kb);
      y[c] = __builtin_amdgcn_wmma_f32_16x16x4_f32(
          false, a, false, b, (short)0, y[c], false, false);
    }
  }
#pragma unroll
  for (int c = 0; c < 4; ++c)
#pragma unroll
    for (int j = 0; j < 8; ++j) y[c][j] = fmaxf(y[c][j], 0.0f);
}

// Coalesced global(16x64 tile) -> LDS, zero-filling out-of-range rows.
__device__ __forceinline__ void load_tile(const float* __restrict__ src,
                                          float* myTile, int lane,
                                          int row0, int n_rows) {
  const f4* s4 = (const f4*)src;
#pragma unroll
  for (int i = 0; i < 8; ++i) {
    int idx = i * 32 + lane;            // float4 index within tile [0,256)
    int r   = row0 + (idx >> 4);        // 16 float4 per row
    f4 v = {0.0f, 0.0f, 0.0f, 0.0f};
    if (r < n_rows) v = s4[(long)row0 * 16 + idx];
    ((f4*)myTile)[idx] = v;
  }
}

// ---------------- Kernel 0: zero the stats accumulators ----------------
__global__ void lgru_zero_kernel(float* __restrict__ ws) {
  if (threadIdx.x < 64) ws[threadIdx.x] = 0.0f;
}

// ---------------- Kernel A: GEMM1+GEMM2+relu, group-norm partial sums ----------------
__global__ void __launch_bounds__(32 * WAVES)
lgru_stats_kernel(const float* __restrict__ h_lv, const float* __restrict__ Wh,
                  const float* __restrict__ bh, const float* __restrict__ Wc,
                  float* __restrict__ ws, int n_prev) {
  __shared__ float sW1[CCH * CCH];
  __shared__ float sW2[CCH * CCH];
  __shared__ float sTile[WAVES][16 * CCH];

  stage_weights(Wh, Wc, sW1, sW2);
  __syncthreads();

  int lane = threadIdx.x & 31;
  int wid  = threadIdx.x >> 5;
  int row0 = (blockIdx.x * WAVES + wid) * 16;
  float* myTile = sTile[wid];

  load_tile(h_lv, myTile, lane, row0, n_prev);
  lds_fence();

  v8f y[4];
  hidden_forward(myTile, sW1, sW2, bh, lane, y);

  int n  = lane & 15;
  int mb = (lane >> 4) << 3;
#pragma unroll
  for (int c = 0; c < 4; ++c) {
    float s = 0.0f, q = 0.0f;
#pragma unroll
    for (int j = 0; j < 8; ++j) {
      if (row0 + mb + j < n_prev) {      // padded rows contribute exact zeros
        float v = y[c][j];
        s += v;
        q += v * v;
      }
    }
    int g = (c * 16 + n) >> 1;           // 32 groups of 2 channels
    atomicAdd(&ws[g], s);
    atomicAdd(&ws[32 + g], q);
  }
}

// ---------------- Kernel B: fold stats + gamma/beta into per-channel affine ----------------
__global__ void lgru_finalize_kernel(float* __restrict__ ws,
                                     const float* __restrict__ gamma,
                                     const float* __restrict__ beta, int n_total) {
  int c = threadIdx.x;
  if (c < 64) {
    int g = c >> 1;
    float inv_cnt = 1.0f / ((float)n_total * 2.0f);   // stats over all N rows x 2 ch
    float mean = ws[g] * inv_cnt;
    float var  = ws[32 + g] * inv_cnt - mean * mean;
    float rstd = rsqrtf(var + 1e-5f);
    float a = rstd * gamma[c];
    ws[64 + c]  = a;
    ws[128 + c] = beta[c] - mean * a;
  }
}

// ---------------- Kernel C: full pipeline + GEMM3 + sigmoid gate ----------------
__global__ void __launch_bounds__(32 * WAVES)
lgru_main_kernel(const float* __restrict__ h_lv, const float* __restrict__ lv,
                 const float* __restrict__ Wh, const float* __restrict__ bh,
                 const float* __restrict__ Wc, const float* __restrict__ ws,
                 float* __restrict__ out, int n_prev, float inv_scale) {
  __shared__ float sW1[CCH * CCH];
  __shared__ float sW2[CCH * CCH];
  __shared__ float sTile[WAVES][16 * CCH];

  stage_weights(Wh, Wc, sW1, sW2);
  __syncthreads();

  int lane = threadIdx.x & 31;
  int wid  = threadIdx.x >> 5;
  int row0 = (blockIdx.x * WAVES + wid) * 16;
  float* myTile = sTile[wid];

  load_tile(h_lv, myTile, lane, row0, n_prev);
  lds_fence();

  v8f y[4];
  hidden_forward(myTile, sW1, sW2, bh, lane, y);

  int n  = lane & 15;
  int mb = (lane >> 4) << 3;

  // group-norm affine, write Z into the tile for GEMM3
#pragma unroll
  for (int c = 0; c < 4; ++c) {
    float ac = ws[64 + c * 16 + n];
    float bc = ws[128 + c * 16 + n];
#pragma unroll
    for (int j = 0; j < 8; ++j)
      myTile[(mb + j) * CCH + c * 16 + n] = y[c][j] * ac + bc;
  }
  lds_fence();

  // ---- GEMM3: Gp = Z @ Wc^T ----
  v8f gacc[4];
#pragma unroll
  for (int c = 0; c < 4; ++c)
#pragma unroll
    for (int j = 0; j < 8; ++j) gacc[c][j] = 0.0f;
#pragma unroll
  for (int kb = 0; kb < CCH; kb += 4) {
    v2f a = ldA(myTile, lane, kb);
#pragma unroll
    for (int c = 0; c < 4; ++c) {
      v2f b = ldB(sW2, lane, c * 16, kb);
      gacc[c] = __builtin_amdgcn_wmma_f32_16x16x4_f32(
          false, a, false, b, (short)0, gacc[c], false, false);
    }
  }

  // sigmoid gate -> tile (row-major) for coalesced output
#pragma unroll
  for (int c = 0; c < 4; ++c)
#pragma unroll
    for (int j = 0; j < 8; ++j) {
      float t = gacc[c][j] * inv_scale;
      myTile[(mb + j) * CCH + c * 16 + n] = 1.0f / (1.0f + __expf(-t));
    }
  lds_fence();

  // out = gate * lv (coalesced float4)
  const f4* lv4  = (const f4*)lv;
  f4*       out4 = (f4*)out;
#pragma unroll
  for (int i = 0; i < 8; ++i) {
    int idx = i * 32 + lane;
    int r   = row0 + (idx >> 4);
    if (r < n_prev) {
      f4 gv = ((f4*)myTile)[idx];
      f4 xv = lv4[(long)row0 * 16 + idx];
      out4[(long)row0 * 16 + idx] = gv * xv;
    }
  }
}

// ---------------- Kernel D: rows >= n_prev pass through (gate == 1) ----------------
__global__ void lgru_tail_kernel(const float* __restrict__ lv, float* __restrict__ out,
                                 long start4, long end4) {
  const f4* s = (const f4*)lv;
  f4*       d = (f4*)out;
  for (long i = start4 + (long)blockIdx.x * blockDim.x + threadIdx.x; i < end4;
       i += (long)gridDim.x * blockDim.x)
    d[i] = s[i];
}

extern "C" void kernel_launch(void* const* d_in, const int* in_sizes, int n_in,
                              void* d_out, int out_size, void* d_ws, size_t ws_size,
                              hipStream_t stream) {
  const float* lv    = (const float*)d_in[0];
  const float* h_lv  = (const float*)d_in[1];
  const float* Wh    = (const float*)d_in[2];
  const float* bh    = (const float*)d_in[3];
  const float* Wc    = (const float*)d_in[4];
  const float* gamma = (const float*)d_in[5];
  const float* beta  = (const float*)d_in[6];
  float* out = (float*)d_out;
  float* ws  = (float*)d_ws;   // [0:32) sum, [32:64) sumsq, [64:128) a, [128:192) b

  int n_total = in_sizes[0] / CCH;
  int n_prev  = in_sizes[1] / CCH;
  int ntiles  = (n_prev + 15) / 16;
  int blocks  = (ntiles + WAVES - 1) / WAVES;
  float inv_scale = 1.0f / (float)(n_total + CCH);

  lgru_zero_kernel<<<1, 64, 0, stream>>>(ws);
  lgru_stats_kernel<<<blocks, 32 * WAVES, 0, stream>>>(h_lv, Wh, bh, Wc, ws, n_prev);
  lgru_finalize_kernel<<<1, 64, 0, stream>>>(ws, gamma, beta, n_total);
  lgru_main_kernel<<<blocks, 32 * WAVES, 0, stream>>>(h_lv, lv, Wh, bh, Wc, ws, out,
                                                      n_prev, inv_scale);
  long start4 = (long)n_prev * (CCH / 4);
  long end4   = (long)n_total * (CCH / 4);
  long cnt4   = end4 - start4;
  if (cnt4 > 0) {
    int tb = (int)((cnt4 + 255) / 256);
    if (tb > 4096) tb = 4096;
    lgru_tail_kernel<<<tb, 256, 0, stream>>>(lv, out, start4, end4);
  }
}